// lossx_73967926772307
// MI455X (gfx1250) — compile-verified
//
#include <hip/hip_runtime.h>
#include <hip/hip_bf16.h>

typedef __attribute__((ext_vector_type(2))) float v2f;
typedef __attribute__((ext_vector_type(8))) float v8f;

#define NB   64
#define NL   21
#define NC   256
#define LC   (NL * NC)          // 5376
#define F4N  (NB * LC)          // 344064 floats per feature

// ---------------------------------------------------------------------------
// Kernel 1: windowed average pool -> f4[feat][b][l][c]  (the bandwidth kernel)
// feature[b,c,i,j]; sum over i in [left,right), j in [down,upper); divide by
// inclusive size s (faithful reference quirk).
// ---------------------------------------------------------------------------
__global__ void lossx_pool_kernel(const float* __restrict__ f1,
                                  const float* __restrict__ f2,
                                  const float* __restrict__ pre1,
                                  const float* __restrict__ pre2,
                                  float* __restrict__ f4ws) {
    const int l    = blockIdx.x;   // 21
    const int b    = blockIdx.y;   // 64
    const int feat = blockIdx.z;   // 2

    const float* __restrict__ feature = feat ? f2 : f1;
    const float* __restrict__ pre     = feat ? pre2 : pre1;

    const float x = pre[(b * NL + l) * 2 + 0];
    const float y = pre[(b * NL + l) * 2 + 1];
    const int down  = (int)fmaxf(y - 6.0f, 0.0f);
    const int left  = (int)fmaxf(x - 6.0f, 0.0f);
    const int upper = (int)fminf(y + 6.0f, 63.0f);
    const int right = (int)fminf(x + 6.0f, 63.0f);
    const float inv_s = 1.0f / (float)((upper - down + 1) * (right - left + 1));
    const int nrows = right - left;   // along axis 2 (H index i), <= 12
    const int ncols = upper - down;   // along axis 3 (W index j), <= 12

    const int wave = threadIdx.x >> 5;
    const int lane = threadIdx.x & 31;
    const int jj = lane & 15;         // column within window (ncols <= 12 < 16)
    const int rr = lane >> 4;         // row phase (2 rows in flight)

    float* __restrict__ dst = f4ws + (size_t)feat * F4N + ((size_t)b * NL + l) * NC;

    for (int c = wave; c < NC; c += 8) {
        const float* __restrict__ p =
            feature + (((size_t)b * NC + c) * 64 + left) * 64 + down;
        if (c + 8 < NC)
            __builtin_prefetch(p + 8 * 4096, 0, 1);   // next channel's window
        float sum = 0.0f;
        if (jj < ncols) {
            for (int r = rr; r < nrows; r += 2)
                sum += p[r * 64 + jj];
        }
        // full-wave reduction (wave32)
        for (int off = 16; off > 0; off >>= 1)
            sum += __shfl_xor(sum, off, 32);
        if (lane == 0)
            dst[c] = sum * inv_s;
    }
}

// ---------------------------------------------------------------------------
// Kernel 2: mean over batch (a ones-vector GEMM) done with V_WMMA_F32_16X16X4,
// then EMA blend.  One wave per 16-wide tile of the 5376 (l,c) outputs.
//   D(16x16) += A(16x4) * B(4x16),  B = ones
//   A[m,k] = f4[b0+k, tile*16+m]  -> 16 WMMAs cover all 64 batches.
// A layout (ISA 7.12.2): lanes 0-15 and 16-31 both span M=0..15 with different
// K; since B is all-ones the result is invariant to the K permutation.
// D column N=0: lane 0 holds M=r, lane 16 holds M=r+8 in acc VGPR r.
// ---------------------------------------------------------------------------
__global__ void lossx_bmean_wmma_kernel(const float* __restrict__ f4ws,
                                        const float* __restrict__ fea1,
                                        const float* __restrict__ fea2,
                                        float* __restrict__ fcout) {
    const int bid  = blockIdx.x;        // 672 = 2 * 336
    const int feat = bid / (LC / 16);
    const int tile = bid % (LC / 16);
    const int lane = threadIdx.x;       // 32
    const int m  = lane & 15;
    const int k0 = (lane >> 4) << 1;    // 0 or 2

    const size_t fbase = (size_t)feat * F4N;
    const int idx = tile * 16 + m;

    v8f acc = {};
    v2f bones; bones.x = 1.0f; bones.y = 1.0f;
    #pragma unroll
    for (int kb = 0; kb < 16; ++kb) {
        const int b0 = kb * 4 + k0;
        v2f a;
        a.x = f4ws[fbase + (size_t)b0 * LC + idx];
        a.y = f4ws[fbase + (size_t)(b0 + 1) * LC + idx];
        acc = __builtin_amdgcn_wmma_f32_16x16x4_f32(
                false, a, false, bones, (short)0, acc, false, false);
    }

    const float* __restrict__ fea = feat ? fea2 : fea1;
    float* __restrict__ out = fcout + (size_t)feat * LC;
    const int col = tile * 16;
    if (lane == 0) {
        #pragma unroll
        for (int r = 0; r < 8; ++r)
            out[col + r] = 0.999f * (acc[r] * (1.0f / 64.0f)) + 0.001f * fea[col + r];
    } else if (lane == 16) {
        #pragma unroll
        for (int r = 0; r < 8; ++r)
            out[col + 8 + r] = 0.999f * (acc[r] * (1.0f / 64.0f)) + 0.001f * fea[col + 8 + r];
    }
}

// ---------------------------------------------------------------------------
// Kernel 3: log_softmax(fea_c1) / normalize(fea_c2) / masked KL -> scalar.
// Single 256-thread block; trivial cost.
// ---------------------------------------------------------------------------
__global__ void lossx_kl_kernel(const float* __restrict__ fc,
                                float* __restrict__ out) {
    __shared__ float red[NC];
    const int t = threadIdx.x;          // 256 threads, one per channel
    float total = 0.0f;

    for (int l = 0; l < NL; ++l) {
        const float a = fc[l * NC + t];           // fea_c1
        const float b = fc[LC + l * NC + t];      // fea_c2

        red[t] = a; __syncthreads();
        for (int off = 128; off > 0; off >>= 1) {
            if (t < off) red[t] = fmaxf(red[t], red[t + off]);
            __syncthreads();
        }
        const float mx = red[0]; __syncthreads();

        red[t] = expf(a - mx); __syncthreads();
        for (int off = 128; off > 0; off >>= 1) {
            if (t < off) red[t] += red[t + off];
            __syncthreads();
        }
        const float lse = mx + logf(red[0]); __syncthreads();

        red[t] = b; __syncthreads();
        for (int off = 128; off > 0; off >>= 1) {
            if (t < off) red[t] += red[t + off];
            __syncthreads();
        }
        const float S2 = red[0]; __syncthreads();

        const float q = b / S2;
        const float logp = a - lse;
        const float term = (q > 0.0f) ? q * (logf(q) - logp) : 0.0f;
        red[t] = term; __syncthreads();
        for (int off = 128; off > 0; off >>= 1) {
            if (t < off) red[t] += red[t + off];
            __syncthreads();
        }
        if (t == 0) total += red[0];
        __syncthreads();
    }
    if (t == 0) out[0] = total * (1.0f / 21.0f);
}

// ---------------------------------------------------------------------------
extern "C" void kernel_launch(void* const* d_in, const int* in_sizes, int n_in,
                              void* d_out, int out_size, void* d_ws, size_t ws_size,
                              hipStream_t stream) {
    const float* f1   = (const float*)d_in[0];
    const float* f2   = (const float*)d_in[1];
    const float* pre1 = (const float*)d_in[2];
    const float* pre2 = (const float*)d_in[3];
    const float* fea1 = (const float*)d_in[4];
    const float* fea2 = (const float*)d_in[5];
    float* out = (float*)d_out;

    float* ws   = (float*)d_ws;
    float* f4ws = ws;                    // 2 * 344064 floats
    float* fcws = ws + 2 * (size_t)F4N;  // 2 * 5376 floats

    dim3 gA(NL, NB, 2);
    lossx_pool_kernel<<<gA, 256, 0, stream>>>(f1, f2, pre1, pre2, f4ws);

    lossx_bmean_wmma_kernel<<<2 * (LC / 16), 32, 0, stream>>>(f4ws, fea1, fea2, fcws);

    lossx_kl_kernel<<<1, NC, 0, stream>>>(fcws, out);
}